// octahedral_transform_82188494176385
// MI455X (gfx1250) — compile-verified
//
#include <hip/hip_runtime.h>
#include <stdint.h>

// ---------------------------------------------------------------------------
// Octahedral grid_sample == exact signed-permutation gather: with
// align_corners=False, every OCTA rotation maps voxel centers to voxel
// centers exactly, so trilinear weights are exactly {0,1} and the op is a
// pure permutation copy (1 GiB in + 1 GiB out, ~92us floor @ 23.3 TB/s).
//
// Strategy: TDM (tensor_load_to_lds) pulls the strided input box for each
// output tile into LDS; threads gather from LDS and emit coalesced float4
// non-temporal stores.
// ---------------------------------------------------------------------------

typedef unsigned int v4u  __attribute__((ext_vector_type(4)));
typedef int          v8i  __attribute__((ext_vector_type(8)));
typedef int          v4i  __attribute__((ext_vector_type(4)));
typedef float        f4v  __attribute__((ext_vector_type(4)));

// 24 proper octahedral rotations (same table as reference)
__constant__ signed char OCTA_C[24][3][3] = {
 {{1,0,0},{0,1,0},{0,0,1}},   {{1,0,0},{0,0,-1},{0,1,0}},  {{1,0,0},{0,-1,0},{0,0,-1}},
 {{1,0,0},{0,0,1},{0,-1,0}},  {{0,0,1},{0,1,0},{-1,0,0}},  {{-1,0,0},{0,1,0},{0,0,-1}},
 {{0,0,-1},{0,1,0},{1,0,0}},  {{0,-1,0},{1,0,0},{0,0,1}},  {{-1,0,0},{0,-1,0},{0,0,1}},
 {{0,1,0},{-1,0,0},{0,0,1}},  {{0,1,0},{1,0,0},{0,0,-1}},  {{0,-1,0},{-1,0,0},{0,0,-1}},
 {{0,0,1},{0,-1,0},{1,0,0}},  {{0,0,-1},{0,-1,0},{-1,0,0}},{{-1,0,0},{0,0,1},{0,1,0}},
 {{-1,0,0},{0,0,-1},{0,-1,0}},{{0,0,1},{1,0,0},{0,1,0}},   {{0,1,0},{0,0,1},{1,0,0}},
 {{0,-1,0},{0,0,-1},{1,0,0}}, {{0,0,1},{-1,0,0},{0,-1,0}}, {{0,0,-1},{1,0,0},{0,-1,0}},
 {{0,1,0},{0,0,-1},{-1,0,0}}, {{0,-1,0},{0,0,1},{-1,0,0}}, {{0,0,-1},{-1,0,0},{0,1,0}}
};

#define DHW 64
#define CH  8                         // z-planes per tile
#define SLAB_ELEMS (DHW*DHW*DHW)      // 262144
#define TILE_ELEMS (DHW*DHW*CH)       // 32768 -> 128 KB LDS

__global__ __launch_bounds__(1024)
void octahedral_transform_82188494176385_kernel(const float* __restrict__ in,
                                                const int* __restrict__ elem_p,
                                                float* __restrict__ out) {
    extern __shared__ float lds[];

    const int t      = threadIdx.x;
    const int bid    = blockIdx.x;
    const int zchunk = bid & ((DHW / CH) - 1);   // 0..7
    const int slab   = bid >> 3;                 // n*C + c, 0..1023
    const int z0     = zchunk * CH;

    int e = elem_p[0];
    if (e < 0) e = 0;
    if (e > 23) e = 23;

    // Input axis i is fed by output axis map[i], optionally reflected.
    // R_inv = OCTA^T, so R_inv[i][j] = OCTA[e][j][i].
    int map[3] = {0, 0, 0};
    int flip[3] = {0, 0, 0};
    #pragma unroll
    for (int i = 0; i < 3; ++i) {
        #pragma unroll
        for (int j = 0; j < 3; ++j) {
            int v = OCTA_C[e][j][i];
            if (v != 0) { map[i] = j; flip[i] = (v < 0); }
        }
    }

    // Input box = preimage of this output tile: n[i] along input axis i.
    int n[3], lo[3];
    #pragma unroll
    for (int i = 0; i < 3; ++i) {
        if (map[i] == 2) { n[i] = CH;  lo[i] = flip[i] ? (DHW - CH - z0) : z0; }
        else             { n[i] = DHW; lo[i] = 0; }
    }
    int K[3];
    int B0 = 0;
    {
        const int bs[3] = {1, n[0], n[0] * n[1]};   // LDS box strides
        #pragma unroll
        for (int i = 0; i < 3; ++i) {
            K[map[i]] = flip[i] ? -bs[i] : bs[i];
            if (flip[i]) B0 += bs[i] * (n[i] - 1);
        }
    }

    const uint64_t slab_off = (uint64_t)slab * (uint64_t)SLAB_ELEMS;
    const uint64_t gaddr = (uint64_t)(uintptr_t)in +
        4ull * (slab_off + (uint64_t)(lo[0] + DHW * lo[1] + DHW * DHW * lo[2]));

    // ---- Tensor Data Mover: strided input box -> LDS (one wave issues) ----
    if (t == 0) {
        const uint32_t lds_base = (uint32_t)(uintptr_t)&lds[0];

        // D# group 0: count=1 | lds_addr | global_addr[56:0] | type=2
        v4u g0;
        g0.x = 1u;
        g0.y = lds_base;
        g0.z = (uint32_t)(gaddr & 0xFFFFFFFFu);
        g0.w = (uint32_t)((gaddr >> 32) & 0x01FFFFFFu) | (2u << 30);

        // D# group 1: data_size=4B, tensor dims huge (no OOB), 3D tile,
        //             strides: dim0=64 (input y), dim1=4096 (input z)
        const uint32_t td0 = 0x40000000u, td1 = 0x40000000u;
        const uint64_t s0 = (uint64_t)DHW;          // 64
        const uint64_t s1 = (uint64_t)(DHW * DHW);  // 4096
        v8i g1;
        g1[0] = (int)(2u << 16);                                     // data_size=2 (4B)
        g1[1] = (int)((td0 & 0xFFFFu) << 16);                        // tensor_dim0 lo16
        g1[2] = (int)((td0 >> 16) | ((td1 & 0xFFFFu) << 16));        // td0 hi16 | td1 lo16
        g1[3] = (int)((td1 >> 16) | ((uint32_t)n[0] << 16));         // td1 hi16 | tile_dim0
        g1[4] = (int)((uint32_t)n[1] | ((uint32_t)n[2] << 16));      // tile_dim1 | tile_dim2
        g1[5] = (int)(uint32_t)(s0 & 0xFFFFFFFFu);                   // stride0[31:0]
        g1[6] = (int)((uint32_t)((s0 >> 32) & 0xFFFFu) |
                      ((uint32_t)(s1 & 0xFFFFu) << 16));             // s0[47:32] | s1[15:0]
        g1[7] = (int)(uint32_t)((s1 >> 16) & 0xFFFFFFFFu);           // s1[47:16]

        // Groups 2/3: tile_dim3/4 = 0 (unused); tensor dims large.
        v4i g2; g2[0] = 0x40000000; g2[1] = 0x40000000; g2[2] = 0; g2[3] = 0;
        v4i g3; g3[0] = 0; g3[1] = 0; g3[2] = 0; g3[3] = 0;
        v8i g4; g4[0] = 0; g4[1] = 0; g4[2] = 0; g4[3] = 0;
                g4[4] = 0; g4[5] = 0; g4[6] = 0; g4[7] = 0;

        // 6-arg form on this toolchain: (g0, g1, g2, g3, extra, cpol)
        __builtin_amdgcn_tensor_load_to_lds(g0, g1, g2, g3, g4, 0);
        __builtin_amdgcn_s_wait_tensorcnt(0);
    }
    __syncthreads();

    // ---- Gather from LDS, coalesced float4 NT stores to output ----
    const int xo = (t & 15) << 2;   // 0..60 step 4 (lane-fastest)
    const int yo = (t >> 4);        // 0..63
    const int Kx = K[0], Ky = K[1], Kz = K[2];
    const int idx0 = B0 + Kx * xo + Ky * yo;
    float* outp = out + slab_off + (uint64_t)z0 * (DHW * DHW)
                      + (uint64_t)yo * DHW + (uint64_t)xo;

    #pragma unroll
    for (int zp = 0; zp < CH; ++zp) {
        const int idx = idx0 + Kz * zp;
        f4v v;
        v.x = lds[idx];
        v.y = lds[idx + Kx];
        v.z = lds[idx + 2 * Kx];
        v.w = lds[idx + 3 * Kx];
        __builtin_nontemporal_store(v, (f4v*)(outp + (uint64_t)zp * (DHW * DHW)));
    }
}

extern "C" void kernel_launch(void* const* d_in, const int* in_sizes, int n_in,
                              void* d_out, int out_size, void* d_ws, size_t ws_size,
                              hipStream_t stream) {
    const float* in  = (const float*)d_in[0];
    const int* elem  = (const int*)d_in[1];
    float* out       = (float*)d_out;

    const int n_slabs = 32 * 32;                 // N*C
    dim3 grid(n_slabs * (DHW / CH));             // 8192 workgroups
    dim3 block(1024);
    size_t shmem = (size_t)TILE_ELEMS * sizeof(float);   // 128 KB per WG

    hipLaunchKernelGGL(octahedral_transform_82188494176385_kernel,
                       grid, block, shmem, stream, in, elem, out);
}